// CrossDcn_14199161881265
// MI455X (gfx1250) — compile-verified
//
#include <hip/hip_runtime.h>

// CDNA5 / gfx1250, wave32.
typedef __attribute__((ext_vector_type(2))) float v2f;
typedef __attribute__((ext_vector_type(4))) float v4f;
typedef __attribute__((ext_vector_type(8))) float v8f;

#define D_DIM 1024
#define L_DIM 4
#define ROWS_PER_WAVE 16
#define WAVES_PER_BLOCK 8
#define ROWS_PER_BLOCK (ROWS_PER_WAVE * WAVES_PER_BLOCK) // 128
#define WPAD 16  // weight rows padded to 16 WMMA columns (rows 4..15 zero)

// ---------------------------------------------------------------------------
// Prologue: g[0]=0 ; g[i] = (b_0 + ... + b_{i-1}) . w_i   (4 floats -> d_ws)
// ---------------------------------------------------------------------------
__global__ void __launch_bounds__(256) dcn_g_precompute(const float* __restrict__ w,
                                                        const float* __restrict__ b,
                                                        float* __restrict__ g) {
    __shared__ float red0[256];
    __shared__ float red1[256];
    __shared__ float red2[256];
    const int t = threadIdx.x;
    float p1 = 0.f, p2 = 0.f, p3 = 0.f;
    for (int d = t; d < D_DIM; d += 256) {
        const float b0 = b[d], b1 = b[D_DIM + d], b2 = b[2 * D_DIM + d];
        const float c1 = b0, c2 = b0 + b1, c3 = c2 + b2;
        p1 += c1 * w[D_DIM + d];
        p2 += c2 * w[2 * D_DIM + d];
        p3 += c3 * w[3 * D_DIM + d];
    }
    red0[t] = p1; red1[t] = p2; red2[t] = p3;
    __syncthreads();
    for (int s = 128; s > 0; s >>= 1) {
        if (t < s) {
            red0[t] += red0[t + s];
            red1[t] += red1[t + s];
            red2[t] += red2[t + s];
        }
        __syncthreads();
    }
    if (t == 0) {
        g[0] = 0.f;
        g[1] = red0[0];
        g[2] = red1[0];
        g[3] = red2[0];
    }
}

// ---------------------------------------------------------------------------
// Fused cross-stack: per wave, 16 rows.
//  Phase 1: U = X_tile @ W^T via v_wmma_f32_16x16x4_f32
//           (B matrix = zero-padded 16x1024 weights in LDS -> no exec masking)
//  Phase 2: per-row scalar recurrence a_{i+1} = a_i*(1+u_i) + g_i
//  Phase 3: out = a * x + Csum   (coalesced, non-temporal stores)
// ---------------------------------------------------------------------------
__global__ void __launch_bounds__(256) dcn_cross_fused(const float* __restrict__ x,
                                                       const float* __restrict__ w,
                                                       const float* __restrict__ b,
                                                       const float* __restrict__ g4,
                                                       float* __restrict__ out) {
    __shared__ __align__(16) float s_w[WPAD * D_DIM];   // 64 KB: weights, rows 4..15 = 0
    __shared__ __align__(16) float s_csum[D_DIM];       //  4 KB: b0+b1+b2+b3

    const int t = threadIdx.x;

    // Stage weights into LDS, zero-padding columns 4..15 of the WMMA B tile.
    for (int i = t; i < (WPAD * D_DIM) / 4; i += 256) {
        v4f v = {0.f, 0.f, 0.f, 0.f};
        if (i < (L_DIM * D_DIM) / 4) v = ((const v4f*)w)[i];
        ((v4f*)s_w)[i] = v;
    }
    // Csum = sum of the four bias rows (256 x 16B, one v4f per thread).
    {
        const int i = t;
        const v4f b0 = ((const v4f*)b)[i];
        const v4f b1 = ((const v4f*)b)[256 + i];
        const v4f b2 = ((const v4f*)b)[512 + i];
        const v4f b3 = ((const v4f*)b)[768 + i];
        ((v4f*)s_csum)[i] = (b0 + b1) + (b2 + b3);
    }
    __syncthreads();

    const int wv    = t >> 5;          // wave in block: 0..7
    const int ln    = t & 31;          // lane: 0..31
    const int m16   = ln & 15;         // A-matrix row (M) / B-matrix col (N)
    const int khalf = (ln >> 4) << 1;  // lane-half K offset: 0 or 2
    const unsigned rowBase = (unsigned)blockIdx.x * ROWS_PER_BLOCK
                           + (unsigned)wv * ROWS_PER_WAVE;

    const v4f g = *(const v4f*)g4;     // {0, g1, g2, g3} — uniform

    // 32-bit lane offset against uniform base: x spans 128 MB < 2^31,
    // so the compiler can use SADDR + voffset (GVS) addressing.
    const unsigned xoff = (rowBase + (unsigned)m16) * D_DIM + (unsigned)khalf;
    const float* __restrict__ wrow = s_w + m16 * D_DIM + khalf;

    float afin[ROWS_PER_WAVE];

#if __has_builtin(__builtin_amdgcn_wmma_f32_16x16x4_f32)
    // ---- Phase 1: WMMA dot accumulation --------------------------------
    v8f c0 = {0.f, 0.f, 0.f, 0.f, 0.f, 0.f, 0.f, 0.f};
    v8f c1 = c0;
#pragma unroll 4
    for (int k = 0; k < D_DIM; k += 8) {
        // A layout (16x4 f32): VGPR0 = K+khalf, VGPR1 = K+khalf+1
        const v2f a0 = *(const v2f*)(x + xoff + k);
        const v2f a1 = *(const v2f*)(x + xoff + k + 4);
        // B layout (4x16 f32): B[k][n] = W[n][k]; rows n>=4 are zero in LDS.
        const v2f b0v = *(const v2f*)(wrow + k);
        const v2f b1v = *(const v2f*)(wrow + k + 4);
        c0 = __builtin_amdgcn_wmma_f32_16x16x4_f32(false, a0, false, b0v,
                                                   (short)0, c0, false, false);
        c1 = __builtin_amdgcn_wmma_f32_16x16x4_f32(false, a1, false, b1v,
                                                   (short)0, c1, false, false);
    }
    const v8f cs = c0 + c1;

    // ---- Phase 2: broadcast u_i and run recurrence ---------------------
    // C layout: VGPR r holds M=r (lanes 0-15, N=lane) and M=8+r (lanes 16-31).
#pragma unroll
    for (int m = 0; m < ROWS_PER_WAVE; ++m) {
        const int base = (m < 8) ? 0 : 16;
        const int r = m & 7;
        const float u0 = __int_as_float(__builtin_amdgcn_readlane(__float_as_int(cs[r]), base + 0));
        const float u1 = __int_as_float(__builtin_amdgcn_readlane(__float_as_int(cs[r]), base + 1));
        const float u2 = __int_as_float(__builtin_amdgcn_readlane(__float_as_int(cs[r]), base + 2));
        const float u3 = __int_as_float(__builtin_amdgcn_readlane(__float_as_int(cs[r]), base + 3));
        float a = (1.f + u0) + g.x;        // a_0 = 1
        a = a * (1.f + u1) + g.y;
        a = a * (1.f + u2) + g.z;
        a = a * (1.f + u3) + g.w;
        afin[m] = a;
    }
#else
    // ---- Fallback (no WMMA builtin): per-lane partial dots -------------
    {
        float u0 = 0.f, u1 = 0.f, u2 = 0.f, u3 = 0.f;
        for (int k = 0; k < D_DIM; k += 4) {
            const v2f xv = *(const v2f*)(x + xoff + k);
            const int kk = k + khalf;
            u0 += xv.x * s_w[0 * D_DIM + kk] + xv.y * s_w[0 * D_DIM + kk + 1];
            u1 += xv.x * s_w[1 * D_DIM + kk] + xv.y * s_w[1 * D_DIM + kk + 1];
            u2 += xv.x * s_w[2 * D_DIM + kk] + xv.y * s_w[2 * D_DIM + kk + 1];
            u3 += xv.x * s_w[3 * D_DIM + kk] + xv.y * s_w[3 * D_DIM + kk + 1];
        }
        u0 += __shfl_xor(u0, 16);
        u1 += __shfl_xor(u1, 16);
        u2 += __shfl_xor(u2, 16);
        u3 += __shfl_xor(u3, 16);
#pragma unroll
        for (int m = 0; m < ROWS_PER_WAVE; ++m) {
            const float v0 = __int_as_float(__builtin_amdgcn_readlane(__float_as_int(u0), m));
            const float v1 = __int_as_float(__builtin_amdgcn_readlane(__float_as_int(u1), m));
            const float v2 = __int_as_float(__builtin_amdgcn_readlane(__float_as_int(u2), m));
            const float v3 = __int_as_float(__builtin_amdgcn_readlane(__float_as_int(u3), m));
            float a = (1.f + v0) + g.x;
            a = a * (1.f + v1) + g.y;
            a = a * (1.f + v2) + g.z;
            a = a * (1.f + v3) + g.w;
            afin[m] = a;
        }
    }
#endif

    // ---- Phase 3: out = a * x + Csum (coalesced v4f, NT stores) --------
#pragma unroll 1
    for (int m = 0; m < ROWS_PER_WAVE; ++m) {
        const unsigned roff = (rowBase + (unsigned)m) * D_DIM;
        const float am = afin[m];
#pragma unroll
        for (int j = 0; j < 8; ++j) {
            const unsigned d = j * 128 + (unsigned)ln * 4;
            const v4f xv = *(const v4f*)(x + roff + d);
            const v4f csv = *(const v4f*)(s_csum + d);
            v4f o;
            o.x = am * xv.x + csv.x;
            o.y = am * xv.y + csv.y;
            o.z = am * xv.z + csv.z;
            o.w = am * xv.w + csv.w;
            __builtin_nontemporal_store(o, (v4f*)(out + roff + d));
        }
    }
}

// ---------------------------------------------------------------------------
extern "C" void kernel_launch(void* const* d_in, const int* in_sizes, int n_in,
                              void* d_out, int out_size, void* d_ws, size_t ws_size,
                              hipStream_t stream) {
    const float* x = (const float*)d_in[0];   // (B, D) f32
    const float* w = (const float*)d_in[1];   // (L, D) f32
    const float* b = (const float*)d_in[2];   // (L, D) f32
    float* out = (float*)d_out;               // (B, D) f32
    float* g = (float*)d_ws;                  // 4 floats scratch

    const int B = in_sizes[0] / D_DIM;        // 32768

    dcn_g_precompute<<<1, 256, 0, stream>>>(w, b, g);
    dcn_cross_fused<<<B / ROWS_PER_BLOCK, 256, 0, stream>>>(x, w, b, g, out);
}